// Predictor_53051436040796
// MI455X (gfx1250) — compile-verified
//
#include <hip/hip_runtime.h>
#include <hip/hip_bf16.h>
#include <math.h>

// ---------------- problem constants ----------------
#define G_    128
#define NPG_  64
#define N_    (G_ * NPG_)      // 8192
#define D_    256
#define S_    5
#define CDIM_ 25
#define C_    2
#define R_    32
#define RB_   90
#define RA_   180
#define BA_   (RB_ * RA_)      // 16200
#define NPOS_ (R_ * BA_)       // 518400 per graph
#define NW_   (C_ * R_ * CDIM_) // 1600

// output layout (floats), in reference return order
#define OFF_FOCUS  0
#define OFF_TS     (OFF_FOCUS + G_)
#define OFF_PROBS  (OFF_TS + G_)
#define OFF_STOP   (OFF_PROBS + N_ * S_)
#define OFF_COEF   (OFF_STOP + G_)
#define OFF_POS    (OFF_COEF + G_ * NW_)

// workspace layout (floats)
#define WS_H       0                         // [N_, D_]
#define WS_LOGITS  (WS_H + N_ * D_)          // [N_, S_]
#define WS_NODEW   (WS_LOGITS + N_ * S_)     // [N_]
#define WS_X       (WS_NODEW + N_)           // [G_, D_]
#define WS_INT     (WS_X + G_ * D_)          // 2*G_ ints (focus, target)
#define WS_GMAX    (WS_INT + 2 * G_)         // [G_]

typedef float v2f __attribute__((ext_vector_type(2)));
typedef float v8f __attribute__((ext_vector_type(8)));

__device__ __forceinline__ v8f wmma_f32(v2f a, v2f b, v8f c) {
  // D = A(16x4,f32) * B(4x16,f32) + C(16x16,f32)
  return __builtin_amdgcn_wmma_f32_16x16x4_f32(false, a, false, b, (short)0, c,
                                               false, false);
}

__device__ __forceinline__ void atomicMaxF(float* addr, float v) {
  if (v >= 0.f)
    atomicMax((int*)addr, __float_as_int(v));
  else
    atomicMin((unsigned int*)addr, __float_as_uint(v));
}

// ---------------- kernel 1: H = relu(X @ W1 + b1) via f32 WMMA ---------------
// grid (2, 512), block 256 (8 waves). wave -> one 16x16 tile; block: 16 rows x 128 cols.
__global__ void k_gemm1_relu(const float* __restrict__ X,
                             const float* __restrict__ W1,
                             const float* __restrict__ b1,
                             float* __restrict__ H) {
  const int wave = threadIdx.x >> 5;
  const int lane = threadIdx.x & 31;
  const int m    = lane & 15;
  const int hh   = lane >> 4;
  const int ntile = blockIdx.x * 8 + wave;     // 0..15
  const int mtile = blockIdx.y;                // 0..511
  const int row = mtile * 16 + m;
  const int n   = ntile * 16 + m;

  v8f acc = {};
  const float* xrow = X + row * D_;
#pragma unroll 4
  for (int kb = 0; kb < D_; kb += 4) {
    const int k0 = kb + 2 * hh;
    v2f a;
    a.x = xrow[k0];
    a.y = xrow[k0 + 1];
    v2f b;
    b.x = W1[k0 * D_ + n];
    b.y = W1[(k0 + 1) * D_ + n];
    acc = wmma_f32(a, b, acc);
  }
  const float bias = b1[n];
#pragma unroll
  for (int i = 0; i < 8; ++i) {
    const int r = mtile * 16 + i + 8 * hh;
    H[r * D_ + n] = fmaxf(acc[i] + bias, 0.f);
  }
}

// ---------------- kernel 2: species_logits = H @ W2 + b2 ----------------
// one wave per node; lane covers 8 consecutive k.
__global__ void k_logits(const float* __restrict__ H,
                         const float* __restrict__ W2,
                         const float* __restrict__ b2,
                         float* __restrict__ L) {
  const int wave = threadIdx.x >> 5;
  const int lane = threadIdx.x & 31;
  const int node = blockIdx.x * 8 + wave;
  const float* hrow = H + node * D_;
  float s[S_] = {0.f, 0.f, 0.f, 0.f, 0.f};
#pragma unroll
  for (int j = 0; j < 8; ++j) {
    const int k = lane * 8 + j;
    const float hv = hrow[k];
#pragma unroll
    for (int t = 0; t < S_; ++t) s[t] += hv * W2[k * S_ + t];
  }
#pragma unroll
  for (int off = 16; off >= 1; off >>= 1)
#pragma unroll
    for (int t = 0; t < S_; ++t) s[t] += __shfl_xor(s[t], off, 32);
  if (lane == 0) {
#pragma unroll
    for (int t = 0; t < S_; ++t) L[node * S_ + t] = s[t] + b2[t];
  }
}

// ---------------- kernel 3: per-graph segment softmax with stop logit --------
// one block (64 threads) per graph.
__global__ void k_softmax(const float* __restrict__ L,
                          float* __restrict__ probs_out,
                          float* __restrict__ stop_out,
                          float* __restrict__ nodew) {
  __shared__ float red[NPG_];
  const int g = blockIdx.x;
  const int t = threadIdx.x;            // 0..63
  const int node = g * NPG_ + t;
  float l[S_];
  float mx = -INFINITY;
#pragma unroll
  for (int s = 0; s < S_; ++s) {
    l[s] = L[node * S_ + s];
    mx = fmaxf(mx, l[s]);
  }
  red[t] = mx;
  __syncthreads();
  for (int off = 32; off > 0; off >>= 1) {
    if (t < off) red[t] = fmaxf(red[t], red[t + off]);
    __syncthreads();
  }
  const float m = fmaxf(red[0], 0.f);
  __syncthreads();

  float e[S_], ssum = 0.f;
#pragma unroll
  for (int s = 0; s < S_; ++s) {
    e[s] = __expf(l[s] - m);
    ssum += e[s];
  }
  red[t] = ssum;
  __syncthreads();
  for (int off = 32; off > 0; off >>= 1) {
    if (t < off) red[t] += red[t + off];
    __syncthreads();
  }
  const float estop = __expf(-m);
  const float Z = red[0] + estop;
  const float inv = 1.f / Z;
#pragma unroll
  for (int s = 0; s < S_; ++s) probs_out[node * S_ + s] = e[s] * inv;
  nodew[node] = ssum * inv;
  if (t == 0) stop_out[g] = estop * inv;
}

// ---------------- kernel 4: deterministic per-graph sampling + init ----------
__device__ __forceinline__ unsigned long long splitmix(unsigned long long& st) {
  st += 0x9E3779B97F4A7C15ull;
  unsigned long long x = st;
  x = (x ^ (x >> 30)) * 0xBF58476D1CE4E5B9ull;
  x = (x ^ (x >> 27)) * 0x94D049BB133111EBull;
  return x ^ (x >> 31);
}

__global__ void k_sample(const float* __restrict__ nodew,
                         const float* __restrict__ probs,
                         float* __restrict__ out_focus,
                         float* __restrict__ out_ts,
                         int* __restrict__ ws_int,
                         float* __restrict__ gmax) {
  const int g = threadIdx.x;  // single block of G_ threads
  unsigned long long st = 42ull ^ (0x9E3779B97F4A7C15ull * (unsigned long long)(g + 1));
  const float u1 = (float)(splitmix(st) >> 40) * (1.f / 16777216.f);
  const float u2 = (float)(splitmix(st) >> 40) * (1.f / 16777216.f);

  const float* w = nodew + g * NPG_;
  float tot = 0.f;
  for (int i = 0; i < NPG_; ++i) tot += w[i];
  float tgt = u1 * tot, c = 0.f;
  int fi = NPG_ - 1;
  for (int i = 0; i < NPG_; ++i) {
    c += w[i];
    if (c >= tgt) { fi = i; break; }
  }
  const int node = g * NPG_ + fi;

  const float* sp = probs + node * S_;
  float sps = 0.f;
  for (int s = 0; s < S_; ++s) sps += sp[s];
  float tgt2 = u2 * sps, c2 = 0.f;
  int ts = S_ - 1;
  for (int s = 0; s < S_; ++s) {
    c2 += sp[s];
    if (c2 >= tgt2) { ts = s; break; }
  }

  out_focus[g] = (float)node;
  out_ts[g]    = (float)ts;
  ws_int[g]        = node;
  ws_int[G_ + g]   = ts;
  gmax[g] = -INFINITY;
}

// ---------------- kernel 5: x = species_embed[ts] * node_scalars[focus] ------
__global__ void k_buildx(const float* __restrict__ X,
                         const float* __restrict__ SE,
                         const int* __restrict__ ws_int,
                         float* __restrict__ xo) {
  const int g = blockIdx.x;
  const int d = threadIdx.x;
  const int node = ws_int[g];
  const int ts   = ws_int[G_ + g];
  xo[g * D_ + d] = SE[ts * D_ + d] * X[node * D_ + d];
}

// ---------------- kernel 6: coeffs = x @ W_pos via f32 WMMA -----------------
// grid (13, 8), block 256; wave tile ntile = bx*8+wave (<100), mtile = by (<8).
__global__ void k_gemm_wpos(const float* __restrict__ Xf,
                            const float* __restrict__ Wp,
                            float* __restrict__ CO) {
  const int wave = threadIdx.x >> 5;
  const int lane = threadIdx.x & 31;
  const int ntile = blockIdx.x * 8 + wave;
  if (ntile >= NW_ / 16) return;  // wave-uniform
  const int mtile = blockIdx.y;
  const int m  = lane & 15;
  const int hh = lane >> 4;
  const int row = mtile * 16 + m;
  const int n   = ntile * 16 + m;

  v8f acc = {};
  const float* xrow = Xf + row * D_;
#pragma unroll 4
  for (int kb = 0; kb < D_; kb += 4) {
    const int k0 = kb + 2 * hh;
    v2f a;
    a.x = xrow[k0];
    a.y = xrow[k0 + 1];
    v2f b;
    b.x = Wp[k0 * NW_ + n];
    b.y = Wp[(k0 + 1) * NW_ + n];
    acc = wmma_f32(a, b, acc);
  }
#pragma unroll
  for (int i = 0; i < 8; ++i) {
    const int r = mtile * 16 + i + 8 * hh;
    CO[r * NW_ + n] = acc[i];
  }
}

// ---------------- kernel 7: s2grid einsum + fused channel-logsumexp ----------
// per graph: [64x25(pad28)] @ [25x16200] with M ordered m = 2*r_local + c so the
// channel pair sits in adjacent accumulator VGPRs -> logsumexp fuses in regs.
// grid (127, 4, 128): bx = ntile group, by = r-group (8 rows), bz = graph.
__global__ void k_s2grid(const float* __restrict__ CO,
                         const float* __restrict__ SH,
                         float* __restrict__ POS,
                         float* __restrict__ gmax) {
  const int wave = threadIdx.x >> 5;
  const int lane = threadIdx.x & 31;
  const int ntile = blockIdx.x * 8 + wave;   // 0..1015 (1013 valid)
  const int mtile = blockIdx.y;              // 0..3
  const int g     = blockIdx.z;
  const int m  = lane & 15;
  const int hh = lane >> 4;
  const int nn = ntile * 16 + m;             // ba column
  const bool colok = (nn < BA_);

  // A row m: r_local = m>>1 (+mtile*8), channel = m&1
  const float* arow = CO + g * NW_ + (m & 1) * (R_ * CDIM_) + (mtile * 8 + (m >> 1)) * CDIM_;

  v2f afr[7];
#pragma unroll
  for (int s = 0; s < 7; ++s) {
    const int k0 = s * 4 + 2 * hh;
    afr[s].x = (k0 < CDIM_) ? arow[k0] : 0.f;
    afr[s].y = (k0 + 1 < CDIM_) ? arow[k0 + 1] : 0.f;
  }

  v8f acc = {};
#pragma unroll
  for (int s = 0; s < 7; ++s) {
    const int k0 = s * 4 + 2 * hh;
    v2f b;
    b.x = (colok && k0 < CDIM_) ? SH[k0 * BA_ + nn] : 0.f;
    b.y = (colok && k0 + 1 < CDIM_) ? SH[(k0 + 1) * BA_ + nn] : 0.f;
    acc = wmma_f32(afr[s], b, acc);
  }

  if (colok) {
    float lmax = -INFINITY;
#pragma unroll
    for (int j = 0; j < 4; ++j) {
      const float v0 = acc[2 * j];
      const float v1 = acc[2 * j + 1];
      const float mm = fmaxf(v0, v1);
      const float lg = mm + __logf(__expf(v0 - mm) + __expf(v1 - mm));
      const int r = mtile * 8 + j + 4 * hh;
      POS[g * NPOS_ + r * BA_ + nn] = lg;
      lmax = fmaxf(lmax, lg);
    }
    atomicMaxF(&gmax[g], lmax);
  }
}

// ---------------- kernel 8: subtract per-graph max ----------------
// grid (254, 128), block 256; thread handles 8 block-strided elements.
__global__ void k_submax(float* __restrict__ POS, const float* __restrict__ gmax) {
  const int g = blockIdx.y;
  const float mg = gmax[g];
  const int base = g * NPOS_;
  int i = blockIdx.x * 2048 + threadIdx.x;
#pragma unroll
  for (int t = 0; t < 8; ++t) {
    if (i < NPOS_) POS[base + i] -= mg;
    i += 256;
  }
}

// ---------------- launcher ----------------
extern "C" void kernel_launch(void* const* d_in, const int* in_sizes, int n_in,
                              void* d_out, int out_size, void* d_ws, size_t ws_size,
                              hipStream_t stream) {
  (void)in_sizes; (void)n_in; (void)out_size; (void)ws_size;
  const float* X   = (const float*)d_in[0];  // node_scalars [N,D]
  // d_in[1]: n_node (uniform; segment = node / NPG_)
  const float* W1  = (const float*)d_in[2];
  const float* b1  = (const float*)d_in[3];
  const float* W2  = (const float*)d_in[4];
  const float* b2  = (const float*)d_in[5];
  const float* SE  = (const float*)d_in[6];  // species_embed [S,D]
  const float* Wp  = (const float*)d_in[7];  // W_pos [D, 1600]
  const float* SH  = (const float*)d_in[8];  // sh_basis [25, 90, 180]
  // d_in[9]: rng key (deterministic hash sampler used instead)

  float* out = (float*)d_out;
  float* ws  = (float*)d_ws;

  float* H      = ws + WS_H;
  float* L      = ws + WS_LOGITS;
  float* nodew  = ws + WS_NODEW;
  float* xbuf   = ws + WS_X;
  int*   wsint  = (int*)(ws + WS_INT);
  float* gmax   = ws + WS_GMAX;

  float* out_focus = out + OFF_FOCUS;
  float* out_ts    = out + OFF_TS;
  float* out_probs = out + OFF_PROBS;
  float* out_stop  = out + OFF_STOP;
  float* out_coef  = out + OFF_COEF;
  float* out_pos   = out + OFF_POS;

  k_gemm1_relu<<<dim3(2, N_ / 16), 256, 0, stream>>>(X, W1, b1, H);
  k_logits<<<N_ / 8, 256, 0, stream>>>(H, W2, b2, L);
  k_softmax<<<G_, NPG_, 0, stream>>>(L, out_probs, out_stop, nodew);
  k_sample<<<1, G_, 0, stream>>>(nodew, out_probs, out_focus, out_ts, wsint, gmax);
  k_buildx<<<G_, D_, 0, stream>>>(X, SE, wsint, xbuf);
  k_gemm_wpos<<<dim3(13, G_ / 16), 256, 0, stream>>>(xbuf, Wp, out_coef);
  k_s2grid<<<dim3(127, 4, G_), 256, 0, stream>>>(out_coef, SH, out_pos, gmax);
  k_submax<<<dim3(254, G_), 256, 0, stream>>>(out_pos, gmax);
}